// DeformConv2d_59468117180727
// MI455X (gfx1250) — compile-verified
//
#include <hip/hip_runtime.h>
#include <math.h>

#define B_   8
#define CIN  128
#define H_   96
#define W_   96
#define COUT 128
#define K_   9
#define PAD  1

typedef __attribute__((ext_vector_type(2))) float v2f;
typedef __attribute__((ext_vector_type(8))) float v8f;

// ---------------------------------------------------------------------------
// Kernel 0: transpose x  NCHW -> NHWC (one block per (b,h) row, LDS staged)
// ---------------------------------------------------------------------------
__global__ __launch_bounds__(256) void k_transpose(const float* __restrict__ x,
                                                   float* __restrict__ xt) {
  __shared__ float tile[W_ * 130];            // pad 130 -> bank stride 2
  const int bh = blockIdx.x;                  // 0 .. B*H-1
  const int b = bh / H_, h = bh % H_;
  const int t = threadIdx.x;
  for (int idx = t; idx < CIN * W_; idx += 256) {       // coalesced reads
    int c = idx / W_, w = idx % W_;
    tile[w * 130 + c] = x[(((size_t)b * CIN + c) * H_ + h) * W_ + w];
  }
  __syncthreads();
  for (int idx = t; idx < W_ * CIN; idx += 256) {       // coalesced writes
    int w = idx / CIN, c = idx % CIN;
    xt[(((size_t)b * H_ + h) * W_ + w) * CIN + c] = tile[w * 130 + c];
  }
}

// ---------------------------------------------------------------------------
// Kernel 1: offset conv (18 channels), layout out: off[b][h][w][18]
// ---------------------------------------------------------------------------
__global__ __launch_bounds__(256) void k_offsets(const float* __restrict__ x,
                                                 const float* __restrict__ ow,
                                                 const float* __restrict__ ob,
                                                 float* __restrict__ off) {
  const int pid = blockIdx.x * 256 + threadIdx.x;       // b*H*W pixels
  if (pid >= B_ * H_ * W_) return;
  const int w = pid % W_;
  const int h = (pid / W_) % H_;
  const int b = pid / (H_ * W_);
  float acc[18];
#pragma unroll
  for (int j = 0; j < 18; ++j) acc[j] = ob[j];
  for (int c = 0; c < CIN; ++c) {
    const float* xc = x + (((size_t)b * CIN + c) * H_) * W_;
    const float* wc = ow + c * 9;
#pragma unroll
    for (int tap = 0; tap < 9; ++tap) {
      const int yy = h + tap / 3 - PAD, xx = w + tap % 3 - PAD;
      float xv = 0.f;
      if (yy >= 0 && yy < H_ && xx >= 0 && xx < W_) xv = xc[yy * W_ + xx];
#pragma unroll
      for (int j = 0; j < 18; ++j)
        acc[j] = fmaf(xv, wc[j * (CIN * 9) + tap], acc[j]);   // uniform -> SMEM
    }
  }
  float* op = off + (size_t)pid * 18;
#pragma unroll
  for (int j = 0; j < 18; ++j) op[j] = acc[j];
}

// ---------------------------------------------------------------------------
// Kernel 2: pack conv_w into per-lane WMMA B-fragments for f32 16x16x4.
// Assumed B layout (mirrors documented A layout): VGPR j, lane l ->
//   row K = j + 2*(l>=16), col N = (l&15) within the 16-wide N tile.
// wp[(((k*32+cg)*8+nt)*32+lane)*2+j] = conv_w[n][cg*4+row][k]
// ---------------------------------------------------------------------------
__global__ __launch_bounds__(256) void k_pack(const float* __restrict__ cw,
                                              float* __restrict__ wp) {
  const int i = blockIdx.x * 256 + threadIdx.x;   // over 9*32*8*32*2
  if (i >= (K_ << 14)) return;
  const int j    = i & 1;
  const int lane = (i >> 1) & 31;
  const int nt   = (i >> 6) & 7;
  const int cg   = (i >> 9) & 31;
  const int k    = i >> 14;
  const int row  = j + 2 * (lane >> 4);
  const int c    = cg * 4 + row;
  const int n    = nt * 16 + (lane & 15);
  wp[i] = cw[((size_t)n * CIN + c) * K_ + k];
}

// ---------------------------------------------------------------------------
// Kernel 3: fused bilinear gather + fp32 WMMA GEMM.
// Block: 256 thr = 8 waves. Tile: M=32 pixels (one (b,h) row) x N=128 couts.
// Waves: wave_m in {0,1} (16 px each), wave_n in {0..3} (32 couts = 2 WMMA N).
// ---------------------------------------------------------------------------
__global__ __launch_bounds__(256) void k_deform_wmma(
    const float* __restrict__ xt, const float* __restrict__ off,
    const float* __restrict__ wp, const float* __restrict__ cb,
    float* __restrict__ out) {
  __shared__ float apatch[32 * 132];        // 32 px x 128 cin, pad 132 (16B ok,
                                            // bank stride 4 -> conflict free)
  const int blk = blockIdx.x;               // b * H * 3
  const int w0  = (blk % 3) * 32;
  const int h   = (blk / 3) % H_;
  const int b   = blk / (3 * H_);

  const int t      = threadIdx.x;
  const int lane   = t & 31;
  const int wv     = t >> 5;
  const int wave_m = wv & 1;
  const int wave_n = wv >> 1;

  v8f acc0 = {}; v8f acc1 = {};

  // gather mapping: 8 threads per pixel, 16 channels each
  const int gpx = t >> 3;
  const int gc0 = (t & 7) * 16;
  const float* offp =
      off + ((((size_t)b * H_ + h) * W_) + (w0 + gpx)) * 18;

  const int arow  = wave_m * 16 + (lane & 15);
  const int khalf = lane >> 4;

  for (int k = 0; k < K_; ++k) {
    // ---------- gather phase: A tile for tap k ----------
    {
      const int ky = k / 3, kx = k % 3;
      const float oy = offp[2 * k + 0];
      const float ox = offp[2 * k + 1];
      const float ys = (float)(h - PAD + ky) + oy;
      const float xs = (float)(w0 + gpx - PAD + kx) + ox;
      const float y0f = floorf(ys), x0f = floorf(xs);
      const float wy = ys - y0f, wx = xs - x0f;
      const int y0 = (int)y0f, x0 = (int)x0f;
      float r[16];
#pragma unroll
      for (int i = 0; i < 16; ++i) r[i] = 0.f;
#pragma unroll
      for (int cor = 0; cor < 4; ++cor) {
        const int dy = cor >> 1, dx = cor & 1;
        const int yi = y0 + dy, xi = x0 + dx;
        const float wgt = (dy ? wy : 1.f - wy) * (dx ? wx : 1.f - wx);
        if (yi >= 0 && yi < H_ && xi >= 0 && xi < W_) {
          const float* p =
              xt + ((((size_t)b * H_ + yi) * W_) + xi) * CIN + gc0;
#pragma unroll
          for (int i = 0; i < 16; i += 4) {
            const float4 v = *(const float4*)(p + i);   // 512B/corner/px
            r[i + 0] = fmaf(wgt, v.x, r[i + 0]);
            r[i + 1] = fmaf(wgt, v.y, r[i + 1]);
            r[i + 2] = fmaf(wgt, v.z, r[i + 2]);
            r[i + 3] = fmaf(wgt, v.w, r[i + 3]);
          }
        }
      }
      float* dst = &apatch[gpx * 132 + gc0];
#pragma unroll
      for (int i = 0; i < 16; i += 4)
        *(float4*)(dst + i) = make_float4(r[i], r[i + 1], r[i + 2], r[i + 3]);
    }
    __syncthreads();

    // ---------- MMA phase: 32 K-steps of v_wmma_f32_16x16x4_f32 ----------
    const float* abase = &apatch[arow * 132 + 2 * khalf];
#pragma unroll 4
    for (int cg = 0; cg < 32; ++cg) {
      v2f a;                       // A frag: K = 2*khalf + {0,1}
      a.x = abase[cg * 4 + 0];
      a.y = abase[cg * 4 + 1];
      const float* wpk =
          wp + (((size_t)(k * 32 + cg) * 8 + wave_n * 2) * 32 + lane) * 2;
      v2f b0; b0.x = wpk[0];  b0.y = wpk[1];
      v2f b1; b1.x = wpk[64]; b1.y = wpk[65];
      acc0 = __builtin_amdgcn_wmma_f32_16x16x4_f32(
          false, a, false, b0, (short)0, acc0, false, false);
      acc1 = __builtin_amdgcn_wmma_f32_16x16x4_f32(
          false, a, false, b1, (short)0, acc1, false, false);
    }
    __syncthreads();
  }

  // ---------- epilogue: C layout VGPR v -> M = v + 8*(lane>=16) ----------
  {
    const int n0 = wave_n * 32 + (lane & 15);
    const int n1 = n0 + 16;
    const float bias0 = cb[n0], bias1 = cb[n1];
    const int mbase = wave_m * 16 + (lane >> 4) * 8;
#pragma unroll
    for (int v = 0; v < 8; ++v) {
      const int px = w0 + mbase + v;
      out[(((size_t)b * COUT + n0) * H_ + h) * W_ + px] = acc0[v] + bias0;
      out[(((size_t)b * COUT + n1) * H_ + h) * W_ + px] = acc1[v] + bias1;
    }
  }
}

// ---------------------------------------------------------------------------
extern "C" void kernel_launch(void* const* d_in, const int* in_sizes, int n_in,
                              void* d_out, int out_size, void* d_ws,
                              size_t ws_size, hipStream_t stream) {
  const float* x  = (const float*)d_in[0];   // (8,128,96,96)
  const float* ow = (const float*)d_in[1];   // (18,128,3,3)
  const float* ob = (const float*)d_in[2];   // (18,)
  // d_in[3], d_in[4]: mod_w / mod_b -> modulator is dead code in reference
  const float* cw = (const float*)d_in[5];   // (128,128,3,3)
  const float* cb = (const float*)d_in[6];   // (128,)
  float* out = (float*)d_out;                // (8,128,96,96)

  float* xt  = (float*)d_ws;                             // 9,437,184 f
  float* off = xt + (size_t)B_ * H_ * W_ * CIN;          // 1,327,104 f
  float* wpk = off + (size_t)B_ * H_ * W_ * 18;          //   147,456 f

  k_transpose<<<B_ * H_, 256, 0, stream>>>(x, xt);
  k_offsets<<<(B_ * H_ * W_ + 255) / 256, 256, 0, stream>>>(x, ow, ob, off);
  k_pack<<<((K_ << 14) + 255) / 256, 256, 0, stream>>>(cw, wpk);
  k_deform_wmma<<<B_ * H_ * 3, 256, 0, stream>>>(xt, off, wpk, cb, out);
}